// SS2D_14113262535151
// MI455X (gfx1250) — compile-verified
//
#include <hip/hip_runtime.h>

typedef __attribute__((ext_vector_type(2))) float v2f;
typedef __attribute__((ext_vector_type(8))) float v8f;

#define D_MODEL 384
#define D_INNER 768
#define NPROJ   1536
#define DT_RANK 24
#define D_STATE 16
#define NDBC    64          // padded 56 -> 64
#define BATCH   8
#define LSEQ    1024
#define MROWS   (BATCH * LSEQ)   // 8192
#define NCHUNK  8
#define CHLEN   128

// ---------------------------------------------------------------------------
// fp32 WMMA: D = A(16x4) x B(4x16) + C(16x16), exact fp32 tensor op.
// A frag (2 VGPR): lanes 0-15 hold K=k,k+1 ; lanes 16-31 hold K=k+2,k+3 (M=lane&15)
// B frag (2 VGPR): mirrored with N=lane&15.
// C/D: VGPR r -> M = r + 8*(lane>>4), N = lane&15.
// ---------------------------------------------------------------------------
static __device__ __forceinline__ v8f wmma4(v2f a, v2f b, v8f c) {
    return __builtin_amdgcn_wmma_f32_16x16x4_f32(
        false, a, false, b, (short)0, c, false, false);
}

static __device__ __forceinline__ float sigmoidf_dev(float x) {
    return 1.0f / (1.0f + expf(-x));
}

// ---------------------------------------------------------------------------
// K1: xz = xseq @ in_proj_w.T ; split into xin / z.
// xseq[m, c] = x[b*384*1024 + c*1024 + l]  (m = b*1024 + l)
// Wave computes 16(M) x 64(N): one A frag feeds 4 WMMAs per K-step.
// grid (512, 3), block (32, 8).
// ---------------------------------------------------------------------------
__global__ __launch_bounds__(256) void k_in_proj(const float* __restrict__ x,
                                                 const float* __restrict__ w,
                                                 float* __restrict__ xin,
                                                 float* __restrict__ z) {
    const int lane = threadIdx.x;
    const int r16 = lane & 15;
    const int hi  = lane >> 4;
    const int m0  = blockIdx.x * 16;
    const int nbase = (blockIdx.y * 8 + threadIdx.y) * 64;
    const int m   = m0 + r16;
    const int b   = m >> 10;
    const int l   = m & 1023;
    const float* ax  = x + (size_t)b * (D_MODEL * LSEQ) + l;   // + c*1024
    const float* bw0 = w + (size_t)(nbase + r16) * D_MODEL;    // tile tb: +tb*16*384

    v8f acc[4] = {};
    for (int k = 0; k < D_MODEL; k += 4) {
        const int ka = k + 2 * hi;
        v2f a;
        a.x = ax[(size_t)ka * LSEQ];
        a.y = ax[(size_t)(ka + 1) * LSEQ];
        #pragma unroll
        for (int tb = 0; tb < 4; ++tb) {
            v2f bb = *(const v2f*)(bw0 + (size_t)tb * 16 * D_MODEL + ka);
            acc[tb] = wmma4(a, bb, acc[tb]);
        }
    }

    // nbase is 64-aligned and 768 % 64 == 0 -> whole wave span is in xin or z
    const bool toXin = (nbase < D_INNER);
    float* base = toXin ? xin : z;
    const int ncol0 = toXin ? nbase : (nbase - D_INNER);
    #pragma unroll
    for (int tb = 0; tb < 4; ++tb) {
        const int n = ncol0 + tb * 16 + r16;
        #pragma unroll
        for (int r = 0; r < 8; ++r) {
            const int mm = m0 + r + 8 * hi;
            base[(size_t)mm * D_INNER + n] = acc[tb][r];
        }
    }
}

// ---------------------------------------------------------------------------
// K2: depthwise causal conv (kernel 3, left-pad 2 per batch) + SiLU
// ---------------------------------------------------------------------------
__global__ __launch_bounds__(256) void k_conv_silu(const float* __restrict__ xin,
                                                   const float* __restrict__ cw,
                                                   const float* __restrict__ cb,
                                                   float* __restrict__ u) {
    const int t = blockIdx.x * blockDim.x + threadIdx.x;
    const int d = t % D_INNER;
    const int m = t / D_INNER;
    const int l = m & 1023;
    float acc = cb[d];
    #pragma unroll
    for (int k = 0; k < 3; ++k) {
        const int ll = l + k - 2;
        if (ll >= 0)
            acc += xin[(size_t)(m + k - 2) * D_INNER + d] * cw[d * 3 + k];
    }
    u[t] = acc * sigmoidf_dev(acc);
}

// ---------------------------------------------------------------------------
// K3: dbc = u @ x_proj_w.T  (N = 56 padded to 64; one wave covers all 4 N-tiles)
// grid (64), block (32, 8): wave ty handles M-tile blockIdx.x*8 + ty.
// ---------------------------------------------------------------------------
__global__ __launch_bounds__(256) void k_x_proj(const float* __restrict__ u,
                                                const float* __restrict__ xw,
                                                float* __restrict__ dbc) {
    const int lane = threadIdx.x;
    const int r16 = lane & 15;
    const int hi  = lane >> 4;
    const int m0  = (blockIdx.x * 8 + threadIdx.y) * 16;
    const float* au = u + (size_t)(m0 + r16) * D_INNER;
    // tile 3 rows 48..63: only n < 56 exist; clamp invalid lanes to row 0
    const float* bw[4];
    #pragma unroll
    for (int tb = 0; tb < 3; ++tb) bw[tb] = xw + (size_t)(tb * 16 + r16) * D_INNER;
    bw[3] = xw + (size_t)((r16 < 8) ? (48 + r16) : 0) * D_INNER;

    v8f acc[4] = {};
    for (int k = 0; k < D_INNER; k += 4) {
        const int ka = k + 2 * hi;
        v2f a = *(const v2f*)(au + ka);
        #pragma unroll
        for (int tb = 0; tb < 4; ++tb) {
            v2f bb = *(const v2f*)(bw[tb] + ka);
            acc[tb] = wmma4(a, bb, acc[tb]);
        }
    }
    #pragma unroll
    for (int tb = 0; tb < 4; ++tb) {
        const int n = tb * 16 + r16;
        if (n < DT_RANK + 2 * D_STATE) {
            #pragma unroll
            for (int r = 0; r < 8; ++r)
                dbc[(size_t)(m0 + r + 8 * hi) * NDBC + n] = acc[tb][r];
        }
    }
}

// ---------------------------------------------------------------------------
// K4: dt = softplus(dbc[:, :24] @ dt_proj_w.T + dt_proj_b)
// grid (512, 3), block (32, 4): wave covers 16(M) x 64(N).
// ---------------------------------------------------------------------------
__global__ __launch_bounds__(128) void k_dt_proj(const float* __restrict__ dbc,
                                                 const float* __restrict__ dw,
                                                 const float* __restrict__ dbias,
                                                 float* __restrict__ dt) {
    const int lane = threadIdx.x;
    const int r16 = lane & 15;
    const int hi  = lane >> 4;
    const int m0  = blockIdx.x * 16;
    const int nbase = (blockIdx.y * 4 + threadIdx.y) * 64;
    const float* aa = dbc + (size_t)(m0 + r16) * NDBC;

    v8f acc[4] = {};
    #pragma unroll
    for (int k = 0; k < DT_RANK; k += 4) {
        const int ka = k + 2 * hi;
        v2f a = *(const v2f*)(aa + ka);
        #pragma unroll
        for (int tb = 0; tb < 4; ++tb) {
            v2f bb = *(const v2f*)(dw + (size_t)(nbase + tb * 16 + r16) * DT_RANK + ka);
            acc[tb] = wmma4(a, bb, acc[tb]);
        }
    }
    #pragma unroll
    for (int tb = 0; tb < 4; ++tb) {
        const int n = nbase + tb * 16 + r16;
        const float bias = dbias[n];
        #pragma unroll
        for (int r = 0; r < 8; ++r) {
            const int mm = m0 + r + 8 * hi;
            const float v = acc[tb][r] + bias;
            const float sp = (v > 20.0f) ? v : log1pf(expf(v));
            dt[(size_t)mm * D_INNER + n] = sp;
        }
    }
}

// ---------------------------------------------------------------------------
// K5a: chunked selective scan, pass 1. thread t = ((b*8 + c) * 768) + d.
// 768 = 3*256 -> every 256-thread block shares one (b, c): stage the shared
// B rows for 32 steps at a time in LDS.
// ---------------------------------------------------------------------------
__global__ __launch_bounds__(256) void k_scan_pass1(const float* __restrict__ dt,
                                                    const float* __restrict__ u,
                                                    const float* __restrict__ dbc,
                                                    const float* __restrict__ A_log,
                                                    float* __restrict__ chP,
                                                    float* __restrict__ chH) {
    __shared__ float sB[32 * D_STATE];   // 32 steps x 16 B values
    const int tid = threadIdx.x;
    const int t  = blockIdx.x * blockDim.x + tid;
    const int d  = t % D_INNER;
    const int bc = t / D_INNER;
    const int b  = bc >> 3;
    const int c  = bc & 7;

    float Arow[D_STATE], P[D_STATE], H[D_STATE];
    #pragma unroll
    for (int n = 0; n < D_STATE; ++n) {
        Arow[n] = -expf(A_log[d * D_STATE + n]);
        P[n] = 1.0f;
        H[n] = 0.0f;
    }

    const int row0 = b * LSEQ + c * CHLEN;
    for (int tile = 0; tile < CHLEN / 32; ++tile) {
        __syncthreads();
        #pragma unroll
        for (int i = 0; i < 2; ++i) {                 // 512 elements / 256 threads
            const int e  = tid + 256 * i;
            const int li = e >> 4;
            const int j  = e & 15;
            sB[e] = dbc[(size_t)(row0 + tile * 32 + li) * NDBC + DT_RANK + j];
        }
        __syncthreads();
        for (int li = 0; li < 32; ++li) {
            const int row = row0 + tile * 32 + li;
            const float dtv = dt[(size_t)row * D_INNER + d];
            const float ut  = u[(size_t)row * D_INNER + d];
            const float du  = dtv * ut;
            const float* bp = sB + li * D_STATE;
            #pragma unroll
            for (int n = 0; n < D_STATE; ++n) {
                const float dA = expf(dtv * Arow[n]);
                H[n] = dA * H[n] + du * bp[n];
                P[n] *= dA;
            }
        }
    }
    #pragma unroll
    for (int n = 0; n < D_STATE; ++n) {
        chP[(size_t)t * D_STATE + n] = P[n];
        chH[(size_t)t * D_STATE + n] = H[n];
    }
}

// ---------------------------------------------------------------------------
// K5b: combine chunk boundary states; one thread per (b, d).
// ---------------------------------------------------------------------------
__global__ __launch_bounds__(256) void k_scan_combine(const float* __restrict__ chP,
                                                      const float* __restrict__ chH,
                                                      float* __restrict__ h0) {
    const int t = blockIdx.x * blockDim.x + threadIdx.x;  // b*768 + d
    const int d = t % D_INNER;
    const int b = t / D_INNER;
    float h[D_STATE];
    #pragma unroll
    for (int n = 0; n < D_STATE; ++n) h[n] = 0.0f;
    for (int c = 0; c < NCHUNK; ++c) {
        const size_t base = ((size_t)((b * NCHUNK + c) * D_INNER) + d) * D_STATE;
        #pragma unroll
        for (int n = 0; n < D_STATE; ++n) {
            h0[base + n] = h[n];
            h[n] = chP[base + n] * h[n] + chH[base + n];
        }
    }
}

// ---------------------------------------------------------------------------
// K5c: pass 2 — re-scan each chunk from its true initial state, produce
// y = (scan_y + u*D) * silu(z). yf aliases z (element read-before-write).
// B and C rows staged in LDS per 32-step tile.
// ---------------------------------------------------------------------------
__global__ __launch_bounds__(256) void k_scan_pass2(const float* __restrict__ dt,
                                                    const float* __restrict__ u,
                                                    const float* __restrict__ dbc,
                                                    const float* __restrict__ A_log,
                                                    const float* __restrict__ Dvec,
                                                    const float* __restrict__ z,
                                                    const float* __restrict__ h0,
                                                    float* __restrict__ yf) {
    __shared__ float sBC[32 * 2 * D_STATE];   // 32 steps x (16 B + 16 C)
    const int tid = threadIdx.x;
    const int t  = blockIdx.x * blockDim.x + tid;
    const int d  = t % D_INNER;
    const int bc = t / D_INNER;
    const int b  = bc >> 3;
    const int c  = bc & 7;

    float Arow[D_STATE], H[D_STATE];
    #pragma unroll
    for (int n = 0; n < D_STATE; ++n) {
        Arow[n] = -expf(A_log[d * D_STATE + n]);
        H[n] = h0[(size_t)t * D_STATE + n];
    }
    const float Dd = Dvec[d];

    const int row0 = b * LSEQ + c * CHLEN;
    for (int tile = 0; tile < CHLEN / 32; ++tile) {
        __syncthreads();
        #pragma unroll
        for (int i = 0; i < 4; ++i) {                 // 1024 elements / 256 threads
            const int e  = tid + 256 * i;
            const int li = e >> 5;
            const int j  = e & 31;
            sBC[e] = dbc[(size_t)(row0 + tile * 32 + li) * NDBC + DT_RANK + j];
        }
        __syncthreads();
        for (int li = 0; li < 32; ++li) {
            const int row = row0 + tile * 32 + li;
            const float dtv = dt[(size_t)row * D_INNER + d];
            const float ut  = u[(size_t)row * D_INNER + d];
            const float du  = dtv * ut;
            const float* bp = sBC + li * 2 * D_STATE;
            float y = 0.0f;
            #pragma unroll
            for (int n = 0; n < D_STATE; ++n) {
                const float dA = expf(dtv * Arow[n]);
                H[n] = dA * H[n] + du * bp[n];
                y += H[n] * bp[D_STATE + n];
            }
            y += ut * Dd;
            const float zv = z[(size_t)row * D_INNER + d];
            yf[(size_t)row * D_INNER + d] = y * (zv * sigmoidf_dev(zv));
        }
    }
}

// ---------------------------------------------------------------------------
// K6: out = yf @ out_proj_w.T, written back in (B, C, H, W) layout.
// grid (512), block (32, 6): wave ty covers N in [ty*64, ty*64+64).
// ---------------------------------------------------------------------------
__global__ __launch_bounds__(192) void k_out_proj(const float* __restrict__ yf,
                                                  const float* __restrict__ ow,
                                                  float* __restrict__ out) {
    const int lane = threadIdx.x;
    const int r16 = lane & 15;
    const int hi  = lane >> 4;
    const int m0  = blockIdx.x * 16;
    const int nbase = threadIdx.y * 64;
    const float* aa  = yf + (size_t)(m0 + r16) * D_INNER;
    const float* bw0 = ow + (size_t)(nbase + r16) * D_INNER;

    v8f acc[4] = {};
    for (int k = 0; k < D_INNER; k += 4) {
        const int ka = k + 2 * hi;
        v2f a = *(const v2f*)(aa + ka);
        #pragma unroll
        for (int tb = 0; tb < 4; ++tb) {
            v2f bb = *(const v2f*)(bw0 + (size_t)tb * 16 * D_INNER + ka);
            acc[tb] = wmma4(a, bb, acc[tb]);
        }
    }
    #pragma unroll
    for (int tb = 0; tb < 4; ++tb) {
        const int n = nbase + tb * 16 + r16;
        #pragma unroll
        for (int r = 0; r < 8; ++r) {
            const int mm = m0 + r + 8 * hi;
            const int bi = mm >> 10;
            const int l  = mm & 1023;
            out[(size_t)bi * (D_MODEL * LSEQ) + (size_t)n * LSEQ + l] = acc[tb][r];
        }
    }
}

// ---------------------------------------------------------------------------
extern "C" void kernel_launch(void* const* d_in, const int* in_sizes, int n_in,
                              void* d_out, int out_size, void* d_ws, size_t ws_size,
                              hipStream_t stream) {
    (void)in_sizes; (void)n_in; (void)out_size; (void)ws_size;
    const float* x    = (const float*)d_in[0];
    const float* w_in = (const float*)d_in[1];
    const float* cw   = (const float*)d_in[2];
    const float* cb   = (const float*)d_in[3];
    const float* xw   = (const float*)d_in[4];
    const float* dw   = (const float*)d_in[5];
    const float* db   = (const float*)d_in[6];
    const float* Al   = (const float*)d_in[7];
    const float* Dv   = (const float*)d_in[8];
    const float* ow   = (const float*)d_in[9];
    float* out = (float*)d_out;

    float* ws  = (float*)d_ws;
    const size_t BIG = (size_t)MROWS * D_INNER;            // 6,291,456 floats
    float* XIN = ws;                                        // aliased by DT later
    float* Z   = ws + BIG;                                  // aliased by YF later
    float* U   = ws + 2 * BIG;
    float* DBC = ws + 3 * BIG;                              // 8192*64
    float* CHP = DBC + (size_t)MROWS * NDBC;
    float* CHH = CHP + (size_t)BATCH * NCHUNK * D_INNER * D_STATE;
    float* H0  = CHH + (size_t)BATCH * NCHUNK * D_INNER * D_STATE;
    float* DT  = XIN;   // xin dead after conv
    float* YF  = Z;     // z consumed element-wise in pass2

    k_in_proj   <<<dim3(MROWS / 16, NPROJ / 512), dim3(32, 8), 0, stream>>>(x, w_in, XIN, Z);
    k_conv_silu <<<(MROWS * D_INNER) / 256, 256, 0, stream>>>(XIN, cw, cb, U);
    k_x_proj    <<<dim3(MROWS / 128), dim3(32, 8), 0, stream>>>(U, xw, DBC);
    k_dt_proj   <<<dim3(MROWS / 16, D_INNER / 256), dim3(32, 4), 0, stream>>>(DBC, dw, db, DT);
    k_scan_pass1<<<(BATCH * NCHUNK * D_INNER) / 256, 256, 0, stream>>>(DT, U, DBC, Al, CHP, CHH);
    k_scan_combine<<<(BATCH * D_INNER) / 256, 256, 0, stream>>>(CHP, CHH, H0);
    k_scan_pass2<<<(BATCH * NCHUNK * D_INNER) / 256, 256, 0, stream>>>(DT, U, DBC, Al, Dv, Z, H0, YF);
    k_out_proj  <<<dim3(MROWS / 16), dim3(32, 6), 0, stream>>>(YF, ow, out);
}